// MSDeformPixelDecoder_11811160064819
// MI455X (gfx1250) — compile-verified
//
#include <hip/hip_runtime.h>
#include <cstdint>
#include <cstddef>

typedef __attribute__((ext_vector_type(2))) float v2f;
typedef __attribute__((ext_vector_type(8))) float v8f;

namespace cfg {
constexpr int C = 256, NH = 8, DH = 32, NL = 3, NP = 4, NLAY = 6, FFN = 1024;
constexpr int N = 5376, B = 2, M = B * N;            // token rows
constexpr int HR = 128, WR = 128, HWR = HR * WR;     // res2 resolution
constexpr int MR = B * HWR;                          // pixel rows
constexpr float EPS = 1e-5f;
}

// ---------------------------------------------------------------------------
// f32 WMMA GEMM:  out[M,Nout] = A[M,K] @ W[Nout,K]^T + bias  (ACT=1 -> ReLU)
// One wave computes a 16 x (16*NT) strip: NT accumulators share each A load,
// giving NT WMMAs per (1 A + NT B) loads instead of 1:2.
// A operand layout (ISA 7.12.2): lane m=l&15, VGPR0/1 hold K=kb, kb+1 with
// kb = 2*(l>>4); B operand mirrored.
// ---------------------------------------------------------------------------
template <int ACT, int NT>
__global__ __launch_bounds__(32) void gemm_f32_wmma(
    const float* __restrict__ A, const float* __restrict__ W,
    const float* __restrict__ bias, float* __restrict__ out,
    int K, int Nout) {
  const int lane  = threadIdx.x;
  const int lm    = lane & 15;
  const int kb    = (lane >> 4) << 1;
  const int mRow  = blockIdx.x * 16 + lm;
  const int nBase = blockIdx.y * (16 * NT);
  const float* ap = A + (size_t)mRow * K + kb;
  const float* bp[NT];
#pragma unroll
  for (int t = 0; t < NT; ++t)
    bp[t] = W + (size_t)(nBase + t * 16 + lm) * K + kb;

  v8f acc[NT];
#pragma unroll
  for (int t = 0; t < NT; ++t)
#pragma unroll
    for (int r = 0; r < 8; ++r) acc[t][r] = 0.f;

#pragma unroll 4
  for (int k = 0; k < K; k += 4) {
    if ((k & 63) == 0) {              // gfx1250 global_prefetch_b8, 512B ahead
      __builtin_prefetch(ap + k + 128, 0, 3);
#pragma unroll
      for (int t = 0; t < NT; ++t) __builtin_prefetch(bp[t] + k + 128, 0, 3);
    }
    v2f a = *reinterpret_cast<const v2f*>(ap + k);
#pragma unroll
    for (int t = 0; t < NT; ++t) {
      v2f b = *reinterpret_cast<const v2f*>(bp[t] + k);
      acc[t] = __builtin_amdgcn_wmma_f32_16x16x4_f32(false, a, false, b,
                                                     (short)0, acc[t], false, false);
    }
  }

  const int mTop = blockIdx.x * 16 + ((lane >> 4) << 3);  // D: M = r + 8*(l>=16)
#pragma unroll
  for (int t = 0; t < NT; ++t) {
    const int nCol = nBase + t * 16 + lm;
    const float bv = bias[nCol];
#pragma unroll
    for (int r = 0; r < 8; ++r) {
      float v = acc[t][r] + bv;
      if (ACT == 1) v = fmaxf(v, 0.f);
      out[(size_t)(mTop + r) * Nout + nCol] = v;
    }
  }
}

// ---------------------------------------------------------------------------
// 3x3 conv (pad 1) as 9 accumulated WMMA GEMMs over pixel-major activations.
// A: [B*HWR, 256] pixel-major, W: [O=256, I=256, 3, 3], out: [B*HWR, 256].
// 16x64 strip per wave; border taps zero-masked with a multiply so loads stay
// unconditional (EXEC all-1s at every WMMA).
// ---------------------------------------------------------------------------
__global__ __launch_bounds__(32) void conv3x3_wmma(
    const float* __restrict__ A, const float* __restrict__ W,
    const float* __restrict__ bias, float* __restrict__ out) {
  using namespace cfg;
  constexpr int NT = 4;
  const int lane = threadIdx.x;
  const int lm   = lane & 15;
  const int kb   = (lane >> 4) << 1;
  const int m    = blockIdx.x * 16 + lm;
  const int b    = m >> 14;               // /16384
  const int hw   = m & (HWR - 1);
  const int y    = hw >> 7, x = hw & (WR - 1);
  const int nBase = blockIdx.y * (16 * NT);
  const float* wb[NT];
#pragma unroll
  for (int t = 0; t < NT; ++t)
    wb[t] = W + (size_t)(nBase + t * 16 + lm) * (256 * 9);

  v8f acc[NT];
#pragma unroll
  for (int t = 0; t < NT; ++t)
#pragma unroll
    for (int r = 0; r < 8; ++r) acc[t][r] = 0.f;

  for (int dy = 0; dy < 3; ++dy) {
    const int yy = y + dy - 1;
    for (int dx = 0; dx < 3; ++dx) {
      const int xx  = x + dx - 1;
      const bool ok = (yy >= 0) && (yy < HR) && (xx >= 0) && (xx < WR);
      const float msk = ok ? 1.f : 0.f;
      const int  shw  = ok ? (yy * WR + xx) : hw;     // safe in-bounds addr
      const float* ap = A + ((size_t)(b * HWR + shw) * 256 + kb);
      const int tap = dy * 3 + dx;
#pragma unroll 4
      for (int k = 0; k < 256; k += 4) {
        v2f a;
        a.x = ap[k] * msk;
        a.y = ap[k + 1] * msk;
#pragma unroll
        for (int t = 0; t < NT; ++t) {
          const float* wp = wb[t] + tap;
          v2f bb;
          bb.x = wp[(size_t)(k + kb) * 9];
          bb.y = wp[(size_t)(k + kb + 1) * 9];
          acc[t] = __builtin_amdgcn_wmma_f32_16x16x4_f32(false, a, false, bb,
                                                         (short)0, acc[t],
                                                         false, false);
        }
      }
    }
  }
  const int mTop = blockIdx.x * 16 + ((lane >> 4) << 3);
#pragma unroll
  for (int t = 0; t < NT; ++t) {
    const int nCol = nBase + t * 16 + lm;
    const float bv = bias[nCol];
#pragma unroll
    for (int r = 0; r < 8; ++r)
      out[(size_t)(mTop + r) * 256 + nCol] = acc[t][r] + bv;
  }
}

// ------------------------- small support kernels ---------------------------
__global__ void pe_init(float* __restrict__ pe) {
  const int n = blockIdx.x, c = threadIdx.x;
  int hw, W;
  if (n < 4096)      { hw = n;        W = 64; }
  else if (n < 5120) { hw = n - 4096; W = 32; }
  else               { hw = n - 5120; W = 16; }
  const float y = (float)(hw / W);
  const float x = (float)(hw % W);
  const int grp = c >> 6, j = c & 63;
  const float dv  = expf(-0.14391156831212787f * (float)j);  // ln(1e4)/64
  const float arg = (grp < 2 ? y : x) * dv;
  pe[(size_t)n * 256 + c] = (grp & 1) ? cosf(arg) : sinf(arg);
}

__global__ void ref_init(float* __restrict__ ref) {
  const int n = blockIdx.x * blockDim.x + threadIdx.x;
  if (n >= cfg::N) return;
  int hw, W, H;
  if (n < 4096)      { hw = n;        W = 64; H = 64; }
  else if (n < 5120) { hw = n - 4096; W = 32; H = 32; }
  else               { hw = n - 5120; W = 16; H = 16; }
  ref[n * 2 + 0] = ((float)(hw % W) + 0.5f) / (float)W;
  ref[n * 2 + 1] = ((float)(hw / W) + 0.5f) / (float)H;
}

__global__ void init_x(const float* __restrict__ r3, const float* __restrict__ r4,
                       const float* __restrict__ r5, const float* __restrict__ lvl,
                       float* __restrict__ x) {
  using namespace cfg;
  const size_t i = (size_t)blockIdx.x * blockDim.x + threadIdx.x;
  if (i >= (size_t)B * N * 256) return;
  const int c = (int)(i % 256);
  const int n = (int)((i / 256) % N);
  const int b = (int)(i / ((size_t)256 * N));
  int li, hw, HW; const float* src;
  if (n < 4096)      { li = 0; hw = n;        HW = 4096; src = r3; }
  else if (n < 5120) { li = 1; hw = n - 4096; HW = 1024; src = r4; }
  else               { li = 2; hw = n - 5120; HW = 256;  src = r5; }
  x[i] = src[((size_t)(b * 256 + c)) * HW + hw] + lvl[li * 256 + c];
}

__global__ void add_pe(const float* __restrict__ x, const float* __restrict__ pe,
                       float* __restrict__ q) {
  using namespace cfg;
  const size_t i = (size_t)blockIdx.x * blockDim.x + threadIdx.x;
  if (i >= (size_t)B * N * 256) return;
  q[i] = x[i] + pe[i % ((size_t)N * 256)];
}

__global__ __launch_bounds__(256) void resid_ln(float* __restrict__ x,
                                                const float* __restrict__ res,
                                                const float* __restrict__ g,
                                                const float* __restrict__ bt) {
  __shared__ float sm[256];
  const int row = blockIdx.x, c = threadIdx.x;
  const float v = x[(size_t)row * 256 + c] + res[(size_t)row * 256 + c];
  sm[c] = v; __syncthreads();
  for (int s = 128; s > 0; s >>= 1) { if (c < s) sm[c] += sm[c + s]; __syncthreads(); }
  const float mu = sm[0] * (1.f / 256.f); __syncthreads();
  const float dv = v - mu;
  sm[c] = dv * dv; __syncthreads();
  for (int s = 128; s > 0; s >>= 1) { if (c < s) sm[c] += sm[c + s]; __syncthreads(); }
  const float var = sm[0] * (1.f / 256.f);
  x[(size_t)row * 256 + c] = dv * rsqrtf(var + cfg::EPS) * g[c] + bt[c];
}

__global__ void softmax12(float* __restrict__ a, int total) {
  const int i = blockIdx.x * blockDim.x + threadIdx.x;
  if (i >= total) return;
  float* p = a + (size_t)i * 12;
  float m = p[0];
#pragma unroll
  for (int j = 1; j < 12; ++j) m = fmaxf(m, p[j]);
  float e[12]; float s = 0.f;
#pragma unroll
  for (int j = 0; j < 12; ++j) { e[j] = expf(p[j] - m); s += e[j]; }
  const float inv = 1.f / s;
#pragma unroll
  for (int j = 0; j < 12; ++j) p[j] = e[j] * inv;
}

// Deformable attention sampling. 32 lanes = DH so value gathers are contiguous.
__global__ void deform_sample(const float* __restrict__ value,  // [B,N,NH,DH]
                              const float* __restrict__ off,    // [B,N,NH,NL,NP,2]
                              const float* __restrict__ aw,     // [B,N,NH,NL*NP]
                              const float* __restrict__ ref,    // [N,2]
                              float* __restrict__ out) {        // [B,N,C]
  using namespace cfg;
  const int d    = threadIdx.x;                                  // 0..31
  const int item = blockIdx.x * blockDim.y + threadIdx.y;        // (b*N+n)*NH+h
  if (item >= B * N * NH) return;
  const int h  = item & 7;
  const int bn = item >> 3;
  const int nn = bn % N;
  const int b  = bn / N;
  const float rx = ref[nn * 2 + 0], ry = ref[nn * 2 + 1];
  const float* offp = off + (size_t)bn * 192 + h * 24;
  const float* awp  = aw  + (size_t)bn * 96  + h * 12;
  const int   Hs[3] = {64, 32, 16}, St[3] = {0, 4096, 5120};
  float acc = 0.f;
#pragma unroll
  for (int li = 0; li < 3; ++li) {
    const int Hh = Hs[li], Ww = Hs[li], st = St[li];
    const float fH = (float)Hh, fW = (float)Ww;
#pragma unroll
    for (int p = 0; p < 4; ++p) {
      const float ox = offp[(li * 4 + p) * 2 + 0];
      const float oy = offp[(li * 4 + p) * 2 + 1];
      const float w  = awp[li * 4 + p];
      const float xf = (rx + ox / fW) * fW - 0.5f;
      const float yf = (ry + oy / fH) * fH - 0.5f;
      const float x0f = floorf(xf), y0f = floorf(yf);
      const float wx = xf - x0f, wy = yf - y0f;
      const int x0 = (int)x0f, y0 = (int)y0f;
      float s = 0.f;
#pragma unroll
      for (int cy = 0; cy < 2; ++cy) {
#pragma unroll
        for (int cx = 0; cx < 2; ++cx) {
          const int xi = x0 + cx, yi = y0 + cy;
          const bool vld = (xi >= 0) && (xi < Ww) && (yi >= 0) && (yi < Hh);
          const float cw = (cx ? wx : 1.f - wx) * (cy ? wy : 1.f - wy);
          int xc = xi < 0 ? 0 : (xi > Ww - 1 ? Ww - 1 : xi);
          int yc = yi < 0 ? 0 : (yi > Hh - 1 ? Hh - 1 : yi);
          const int tok = st + yc * Ww + xc;
          const float g = value[(((size_t)b * N + tok) * 8 + h) * 32 + d];
          s += (vld ? cw : 0.f) * g;
        }
      }
      acc += w * s;
    }
  }
  out[((size_t)b * N + nn) * 256 + h * 32 + d] = acc;
}

__global__ void extract_out(const float* __restrict__ x, float* __restrict__ o,
                            float* __restrict__ o3pm, int st, int HW) {
  using namespace cfg;
  const size_t i = (size_t)blockIdx.x * blockDim.x + threadIdx.x;
  if (i >= (size_t)B * 256 * HW) return;
  const int hw = (int)(i % HW);
  const int c  = (int)((i / HW) % 256);
  const int b  = (int)(i / ((size_t)256 * HW));
  const float v = x[((size_t)(b * N + st + hw)) * 256 + c];
  o[i] = v;
  if (o3pm) o3pm[((size_t)(b * HW + hw)) * 256 + c] = v;
}

__global__ void nchw_to_nhwc(const float* __restrict__ in, float* __restrict__ out) {
  using namespace cfg;
  const size_t i = (size_t)blockIdx.x * blockDim.x + threadIdx.x;
  if (i >= (size_t)B * HWR * 256) return;
  const int c  = (int)(i % 256);
  const int hw = (int)((i / 256) % HWR);
  const int b  = (int)(i / ((size_t)256 * HWR));
  out[i] = in[((size_t)(b * 256 + c)) * HWR + hw];
}

// R[b,hw,c] += bilinear-upsample(o3pm 64x64 -> 128x128), half-pixel centers.
__global__ void upsample_add(float* __restrict__ R, const float* __restrict__ o3pm) {
  using namespace cfg;
  const size_t i = (size_t)blockIdx.x * blockDim.x + threadIdx.x;
  if (i >= (size_t)B * HWR * 256) return;
  const int c  = (int)(i % 256);
  const int hw = (int)((i / 256) % HWR);
  const int b  = (int)(i / ((size_t)256 * HWR));
  const int y = hw >> 7, x = hw & 127;
  const float sy = (float)y * 0.5f - 0.25f;
  const float sx = (float)x * 0.5f - 0.25f;
  const float y0f = floorf(sy), x0f = floorf(sx);
  const float wy = sy - y0f, wx = sx - x0f;
  int y0 = (int)y0f, x0 = (int)x0f;
  int y1 = y0 + 1,   x1 = x0 + 1;
  y0 = y0 < 0 ? 0 : (y0 > 63 ? 63 : y0);
  y1 = y1 < 0 ? 0 : (y1 > 63 ? 63 : y1);
  x0 = x0 < 0 ? 0 : (x0 > 63 ? 63 : x0);
  x1 = x1 < 0 ? 0 : (x1 > 63 ? 63 : x1);
  const size_t base = (size_t)b * 4096;
  const float v00 = o3pm[(base + y0 * 64 + x0) * 256 + c];
  const float v01 = o3pm[(base + y0 * 64 + x1) * 256 + c];
  const float v10 = o3pm[(base + y1 * 64 + x0) * 256 + c];
  const float v11 = o3pm[(base + y1 * 64 + x1) * 256 + c];
  R[i] += (1.f - wy) * ((1.f - wx) * v00 + wx * v01) +
          wy * ((1.f - wx) * v10 + wx * v11);
}

__global__ __launch_bounds__(256) void gn_stats(const float* __restrict__ s,
                                                float* __restrict__ stats) {
  using namespace cfg;
  __shared__ float ssum[256], ssq[256];
  const int b = blockIdx.x >> 5;
  const int g = blockIdx.x & 31;
  const int tid = threadIdx.x;
  const int cnt = 8 * HWR;  // 131072 elements per (b, group)
  float sum = 0.f, sq = 0.f;
  for (int i = tid; i < cnt; i += 256) {
    const int ch = i & 7;
    const int hw = i >> 3;
    const float v = s[((size_t)(b * HWR + hw)) * 256 + g * 8 + ch];
    sum += v; sq += v * v;
  }
  ssum[tid] = sum; ssq[tid] = sq; __syncthreads();
  for (int st = 128; st > 0; st >>= 1) {
    if (tid < st) { ssum[tid] += ssum[tid + st]; ssq[tid] += ssq[tid + st]; }
    __syncthreads();
  }
  if (tid == 0) {
    const float mu  = ssum[0] / (float)cnt;
    const float var = ssq[0] / (float)cnt - mu * mu;
    stats[(b * 32 + g) * 2 + 0] = mu;
    stats[(b * 32 + g) * 2 + 1] = rsqrtf(var + EPS);
  }
}

__global__ void gn_apply_gelu(const float* __restrict__ s,
                              const float* __restrict__ stats,
                              const float* __restrict__ gg,
                              const float* __restrict__ gb,
                              float* __restrict__ out) {  // [B,C,H,W] channel-major
  using namespace cfg;
  const size_t i = (size_t)blockIdx.x * blockDim.x + threadIdx.x;
  if (i >= (size_t)B * 256 * HWR) return;
  const int hw = (int)(i % HWR);
  const int c  = (int)((i / HWR) % 256);
  const int b  = (int)(i / ((size_t)256 * HWR));
  const int g  = c >> 3;
  const float mu = stats[(b * 32 + g) * 2 + 0];
  const float rs = stats[(b * 32 + g) * 2 + 1];
  float v = s[((size_t)(b * HWR + hw)) * 256 + c];
  v = (v - mu) * rs * gg[c] + gb[c];
  out[i] = 0.5f * v * (1.f + erff(v * 0.70710678118654752f));
}

// ---------------------------------------------------------------------------
extern "C" void kernel_launch(void* const* d_in, const int* in_sizes, int n_in,
                              void* d_out, int out_size, void* d_ws, size_t ws_size,
                              hipStream_t stream) {
  using namespace cfg;
  (void)in_sizes; (void)n_in; (void)out_size; (void)ws_size;
  const float* res2  = (const float*)d_in[0];
  const float* res3  = (const float*)d_in[1];
  const float* res4  = (const float*)d_in[2];
  const float* res5  = (const float*)d_in[3];
  const float* lvl   = (const float*)d_in[4];
  const float* off_W = (const float*)d_in[5];
  const float* off_b = (const float*)d_in[6];
  const float* aw_W  = (const float*)d_in[7];
  const float* aw_b  = (const float*)d_in[8];
  const float* val_W = (const float*)d_in[9];
  const float* val_b = (const float*)d_in[10];
  const float* out_W = (const float*)d_in[11];
  const float* out_b = (const float*)d_in[12];
  const float* ln1_g = (const float*)d_in[13];
  const float* ln1_b = (const float*)d_in[14];
  const float* fc1_W = (const float*)d_in[15];
  const float* fc1_b = (const float*)d_in[16];
  const float* fc2_W = (const float*)d_in[17];
  const float* fc2_b = (const float*)d_in[18];
  const float* ln2_g = (const float*)d_in[19];
  const float* ln2_b = (const float*)d_in[20];
  const float* lat_W = (const float*)d_in[21];
  const float* lat_b = (const float*)d_in[22];
  const float* sm_W  = (const float*)d_in[23];
  const float* sm_b  = (const float*)d_in[24];
  const float* gn_g  = (const float*)d_in[25];
  const float* gn_b  = (const float*)d_in[26];

  float* ws = (float*)d_ws;
  // ---- phase A (transformer layers) layout, floats ----
  float* PE   = ws + 0;          // 1,376,256
  float* REF  = ws + 1376256;    //    10,752
  float* X    = ws + 1387008;    // 2,752,512
  float* Q    = ws + 4139520;    // 2,752,512
  float* VALB = ws + 6892032;    // 2,752,512
  float* OFFB = ws + 9644544;    // 2,064,384
  float* AWB  = ws + 11708928;   // 1,032,192
  float* ATT  = ws + 12741120;   // 2,752,512
  float* TMP2 = ws + 15493632;   // 2,752,512
  float* HB   = ws + 18246144;   // 11,010,048  (end 29,256,192 floats ~= 117MB)
  // ---- phase B (FPN) layout, aliases dead phase-A regions ----
  float* T2    = ws + 0;         // 8,388,608 (over PE..VALB, dead)
  float* RB    = ws + 8388608;   // 8,388,608 (over OFFB..TMP2, dead)
  float* O3PM  = ws + 18246144;  // 2,097,152 (inside dead HB)
  float* SPM   = ws + 20343296;  // 8,388,608
  float* STATS = ws + 28731904;  // 128

  // Prologue: positional encodings, reference points, flattened tokens.
  pe_init<<<N, 256, 0, stream>>>(PE);
  ref_init<<<(N + 255) / 256, 256, 0, stream>>>(REF);
  init_x<<<(B * N * 256 + 255) / 256, 256, 0, stream>>>(res3, res4, res5, lvl, X);

  const dim3 wave(32);
  for (int l = 0; l < NLAY; ++l) {
    add_pe<<<(B * N * 256 + 255) / 256, 256, 0, stream>>>(X, PE, Q);
    gemm_f32_wmma<0, 4><<<dim3(M / 16, 4), wave, 0, stream>>>(
        X, val_W + (size_t)l * 65536, val_b + l * 256, VALB, 256, 256);
    gemm_f32_wmma<0, 4><<<dim3(M / 16, 3), wave, 0, stream>>>(
        Q, off_W + (size_t)l * 49152, off_b + l * 192, OFFB, 256, 192);
    gemm_f32_wmma<0, 2><<<dim3(M / 16, 3), wave, 0, stream>>>(
        Q, aw_W + (size_t)l * 24576, aw_b + l * 96, AWB, 256, 96);
    softmax12<<<(B * N * NH + 255) / 256, 256, 0, stream>>>(AWB, B * N * NH);
    deform_sample<<<B * N * NH / 4, dim3(32, 4), 0, stream>>>(VALB, OFFB, AWB, REF, ATT);
    gemm_f32_wmma<0, 4><<<dim3(M / 16, 4), wave, 0, stream>>>(
        ATT, out_W + (size_t)l * 65536, out_b + l * 256, TMP2, 256, 256);
    resid_ln<<<M, 256, 0, stream>>>(X, TMP2, ln1_g + l * 256, ln1_b + l * 256);
    gemm_f32_wmma<1, 4><<<dim3(M / 16, 16), wave, 0, stream>>>(
        X, fc1_W + (size_t)l * 262144, fc1_b + l * 1024, HB, 256, 1024);
    gemm_f32_wmma<0, 4><<<dim3(M / 16, 4), wave, 0, stream>>>(
        HB, fc2_W + (size_t)l * 262144, fc2_b + l * 256, TMP2, 1024, 256);
    resid_ln<<<M, 256, 0, stream>>>(X, TMP2, ln2_g + l * 256, ln2_b + l * 256);
  }

  // Epilogue: emit o3/o4/o5 (channel-major) + pixel-major o3 copy for upsample.
  float* OUT = (float*)d_out;
  float* o3 = OUT + 8388608;    // after s: B*C*128*128
  float* o4 = OUT + 10485760;
  float* o5 = OUT + 11010048;
  extract_out<<<(B * 256 * 4096 + 255) / 256, 256, 0, stream>>>(X, o3, O3PM, 0, 4096);
  extract_out<<<(B * 256 * 1024 + 255) / 256, 256, 0, stream>>>(X, o4, nullptr, 4096, 1024);
  extract_out<<<(B * 256 * 256 + 255) / 256, 256, 0, stream>>>(X, o5, nullptr, 5120, 256);

  // FPN tail.
  nchw_to_nhwc<<<(B * HWR * 256 + 255) / 256, 256, 0, stream>>>(res2, T2);
  gemm_f32_wmma<0, 4><<<dim3(MR / 16, 4), wave, 0, stream>>>(T2, lat_W, lat_b, RB, 256, 256);
  upsample_add<<<(B * HWR * 256 + 255) / 256, 256, 0, stream>>>(RB, O3PM);
  conv3x3_wmma<<<dim3(MR / 16, 4), wave, 0, stream>>>(RB, sm_W, sm_b, SPM);
  gn_stats<<<B * 32, 256, 0, stream>>>(SPM, STATS);
  gn_apply_gelu<<<(B * 256 * HWR + 255) / 256, 256, 0, stream>>>(SPM, STATS, gn_g, gn_b, OUT);
}